// BrainRegistrationLoss_63625645523584
// MI455X (gfx1250) — compile-verified
//
#include <hip/hip_runtime.h>
#include <hip/hip_bf16.h>

// Problem constants (from reference)
#define B_ 2
#define N_ 16384
#define M_ 16384
#define E_ 49152
#define W_CHAMFER 1.0f
#define W_EDGE 0.1f

#define MSPLIT 2
#define MCHUNK (M_ / MSPLIT)               // 8192 columns per WG
#define ROWS_PER_WAVE 32                   // 2 row-tiles per wave
#define ROWS_PER_WG 256                    // 8 waves * 32 rows
#define WGS_PER_BATCH (N_ / ROWS_PER_WG)   // 64
#define EDGE_BLOCKS_PER_BATCH (E_ / 256)   // 192
#define INF_BITS 0x7F800000u

typedef __attribute__((ext_vector_type(2))) float v2f;
typedef __attribute__((ext_vector_type(8))) float v8f;

// Raw v_min_num_f32 accumulator update: avoids the canonicalize (v_max x,x)
// ops the backend inserts around llvm.minnum; HW min_num already quiets NaNs
// per ISA IEEE minimumNumber semantics.
static __device__ __forceinline__ void min_upd(float& acc, float v) {
    asm("v_min_num_f32 %0, %0, %1" : "+v"(acc) : "v"(v));
}

// ---------------- init: rowmin/colmin -> +inf bits ----------------
__global__ void init_ws_kernel(unsigned* __restrict__ rowmin,
                               unsigned* __restrict__ colmin) {
    int i = blockIdx.x * blockDim.x + threadIdx.x;
    if (i < B_ * N_) rowmin[i] = INF_BITS;
    if (i < B_ * M_) colmin[i] = INF_BITS;
}

// ---------------- chamfer: WMMA f32 16x16x4 core ----------------
// d2(n,m) = p2_n + t2_m - 2*dot(p_n, t_m).  The K=4 slot of the f32 WMMA
// carries the affine term, so the matrix op emits the epilogue operand
// directly:
//   row WMMA: A = (-2x,-2y,-2z, 1),   B = (x,y,z, t2_j) -> D = t2_j - 2c
//   col WMMA: A = (-2x,-2y,-2z, p2_i),B = (x,y,z, 1)    -> E = p2_i - 2c
// rowmin_n = p2_n + min_m D   (register partials -> global u32-min)
// colmin_m = t2_m + min_n E   (LDS u32-min -> global u32-min)
__global__ __launch_bounds__(256)
void chamfer_kernel(const float* __restrict__ pred,
                    const float* __restrict__ targ,
                    unsigned* __restrict__ rowmin,
                    unsigned* __restrict__ colmin) {
    __shared__ unsigned cmin[MCHUNK];      // 32 KB of the 320 KB WGP LDS

    const int tid  = threadIdx.x;
    const int lane = tid & 31;
    const int tjl  = lane & 15;            // column-within-tile (both halves)
    const int wave = tid >> 5;             // 0..7
    const int b       = blockIdx.x / WGS_PER_BATCH;
    const int wg_row0 = (blockIdx.x % WGS_PER_BATCH) * ROWS_PER_WG;
    const int base    = wg_row0 + wave * ROWS_PER_WAVE;   // this wave's 32 rows
    const int mbase   = blockIdx.y * MCHUNK;              // column half

    for (int m = tid; m < MCHUNK; m += 256) cmin[m] = INF_BITS;
    __syncthreads();

    const float* __restrict__ pb = pred + (size_t)b * N_ * 3;
    const float* __restrict__ tb = targ + (size_t)b * M_ * 3;

    // A tiles (16x4 f32). ISA layout: v0 = {K0 | K2}, v1 = {K1 | K3} across
    // lane halves. Row/col variants share v0; only the K3 half of v1 differs.
    v2f A0r, A0c, A1r, A1c;
    {
        const int r0 = base + tjl;
        const float x = pb[3 * r0 + 0], y = pb[3 * r0 + 1], z = pb[3 * r0 + 2];
        const float p2 = x * x + y * y + z * z;
        const float mx = -2.0f * x, my = -2.0f * y, mz = -2.0f * z;
        A0r.x = (lane < 16) ? mx : mz;
        A0r.y = (lane < 16) ? my : 1.0f;
        A0c.x = A0r.x;
        A0c.y = (lane < 16) ? my : p2;
    }
    {
        const int r1 = base + 16 + tjl;
        const float x = pb[3 * r1 + 0], y = pb[3 * r1 + 1], z = pb[3 * r1 + 2];
        const float p2 = x * x + y * y + z * z;
        const float mx = -2.0f * x, my = -2.0f * y, mz = -2.0f * z;
        A1r.x = (lane < 16) ? mx : mz;
        A1r.y = (lane < 16) ? my : 1.0f;
        A1c.x = A1r.x;
        A1c.y = (lane < 16) ? my : p2;
    }

    float rowm[16];
#pragma unroll
    for (int r = 0; r < 16; ++r) rowm[r] = __uint_as_float(INF_BITS);

    // One column-tile step: 4 WMMAs (K-slot-fused affine) + pure-min epilogue.
    auto step = [&](int m0, float tx, float ty, float tz) {
        const float t2 = tx * tx + ty * ty + tz * tz;
        const float bx = (lane < 16) ? tx : tz;
        v2f B0, B1;                            // 4x16 B tiles, share K0..K2
        B0.x = bx;  B0.y = (lane < 16) ? ty : t2;     // K3 row = t2_j
        B1.x = bx;  B1.y = (lane < 16) ? ty : 1.0f;   // K3 row = 1

        v8f Z = {};
        v8f D0 = __builtin_amdgcn_wmma_f32_16x16x4_f32(false, A0r, false, B0,
                                                       (short)0, Z, false, false);
        v8f D1 = __builtin_amdgcn_wmma_f32_16x16x4_f32(false, A1r, false, B0,
                                                       (short)0, Z, false, false);
        v8f E0 = __builtin_amdgcn_wmma_f32_16x16x4_f32(false, A0c, false, B1,
                                                       (short)0, Z, false, false);
        v8f E1 = __builtin_amdgcn_wmma_f32_16x16x4_f32(false, A1c, false, B1,
                                                       (short)0, Z, false, false);

        float colm = __uint_as_float(INF_BITS);
#pragma unroll
        for (int r = 0; r < 8; ++r) {
            min_upd(rowm[r],     D0[r]);                       // t2 - 2c
            min_upd(rowm[8 + r], D1[r]);
            colm = fminf(colm, fminf(E0[r], E1[r]));           // p2 - 2c (min3)
        }
        // All 32 lanes atomic-min; the two half-lanes share the same address,
        // so the lane-half combine happens inside the LDS atomic unit.
        const float colval = fmaxf(colm + t2, 0.0f);
        atomicMin(&cmin[m0 + tjl], __float_as_uint(colval));  // ds_min_u32
    };

    // Software pipeline with pointer-bump prefetch; last iteration peeled so
    // the steady-state prefetch is unconditional (constant-offset b96 load).
    // No unrolling: keeps each rowm accumulator coalesced in one register
    // across the loop (the tied inline-asm operand updates it in place).
    const float* __restrict__ tptr = tb + 3 * (mbase + tjl);
    float tx = tptr[0];
    float ty = tptr[1];
    float tz = tptr[2];

    int m0 = 0;
#pragma unroll 1
    for (; m0 < MCHUNK - 16; m0 += 16) {
        const float nx = tptr[48];             // next tile: +16 points = +48 floats
        const float ny = tptr[49];
        const float nz = tptr[50];
        step(m0, tx, ty, tz);
        tx = nx; ty = ny; tz = nz;
        tptr += 48;
    }
    step(m0, tx, ty, tz);                      // peeled tail, no prefetch

    // Row mins: recompute p2 in C/D layout (VGPR r: lanes 0-15 -> row +r,
    // lanes 16-31 -> row +r+8), butterfly reduce, then global u32-min.
#pragma unroll
    for (int r = 0; r < 16; ++r) {
        const int tile = (r >> 3) * 16;        // 0 or 16
        const int rr   = r & 7;
        const int pr   = base + tile + rr + ((lane < 16) ? 0 : 8);
        const float x = pb[3 * pr + 0], y = pb[3 * pr + 1], z = pb[3 * pr + 2];
        const float p2 = x * x + y * y + z * z;
        float v = rowm[r];
        v = fminf(v, __shfl_xor(v, 1, 32));
        v = fminf(v, __shfl_xor(v, 2, 32));
        v = fminf(v, __shfl_xor(v, 4, 32));
        v = fminf(v, __shfl_xor(v, 8, 32));
        v = fmaxf(v + p2, 0.0f);
        if (lane == 0)
            atomicMin(&rowmin[(size_t)b * N_ + base + tile + rr], __float_as_uint(v));
        if (lane == 16)
            atomicMin(&rowmin[(size_t)b * N_ + base + tile + rr + 8], __float_as_uint(v));
    }

    __syncthreads();
    // Flush LDS column mins to global (u32 min on non-negative float bits)
    for (int m = tid; m < MCHUNK; m += 256)
        atomicMin(&colmin[(size_t)b * M_ + mbase + m], cmin[m]);
}

// ---------------- edge lengths: deterministic block partials ----------------
__global__ __launch_bounds__(256)
void edge_kernel(const float* __restrict__ pred,
                 const int* __restrict__ edges,
                 float* __restrict__ epart) {
    __shared__ float s0[256];
    __shared__ float s1[256];
    const int tid = threadIdx.x;
    const int i = blockIdx.x * 256 + tid;   // [0, B*E), E % 256 == 0
    const int b = i / E_;
    const int e = i - b * E_;
    const float* __restrict__ pb = pred + (size_t)b * N_ * 3;
    const int a0 = edges[2 * e + 0];
    const int a1 = edges[2 * e + 1];
    const float dx = pb[3 * a0 + 0] - pb[3 * a1 + 0];
    const float dy = pb[3 * a0 + 1] - pb[3 * a1 + 1];
    const float dz = pb[3 * a0 + 2] - pb[3 * a1 + 2];
    const float len = sqrtf(dx * dx + dy * dy + dz * dz);
    s0[tid] = len;
    s1[tid] = len * len;
    __syncthreads();
    for (int s = 128; s > 0; s >>= 1) {
        if (tid < s) { s0[tid] += s0[tid + s]; s1[tid] += s1[tid + s]; }
        __syncthreads();
    }
    if (tid == 0) {
        epart[2 * blockIdx.x + 0] = s0[0];
        epart[2 * blockIdx.x + 1] = s1[0];
    }
}

// ---------------- finale: sqrt of mins, means, variance, outputs ----------------
__global__ __launch_bounds__(256)
void final_kernel(const unsigned* __restrict__ rowmin,
                  const unsigned* __restrict__ colmin,
                  const float* __restrict__ epart,
                  float* __restrict__ out) {
    __shared__ float sr[256];
    __shared__ float sc[256];
    const int tid = threadIdx.x;
    float rs = 0.0f, cs = 0.0f;
    for (int i = tid; i < B_ * N_; i += 256) rs += sqrtf(__uint_as_float(rowmin[i]));
    for (int i = tid; i < B_ * M_; i += 256) cs += sqrtf(__uint_as_float(colmin[i]));
    sr[tid] = rs;
    sc[tid] = cs;
    __syncthreads();
    for (int s = 128; s > 0; s >>= 1) {
        if (tid < s) { sr[tid] += sr[tid + s]; sc[tid] += sc[tid + s]; }
        __syncthreads();
    }
    if (tid == 0) {
        const float c_loss = sr[0] / (float)(B_ * N_) + sc[0] / (float)(B_ * M_);
        float e_loss = 0.0f;
        for (int b = 0; b < B_; ++b) {
            float sum = 0.0f, sq = 0.0f;
            for (int k = 0; k < EDGE_BLOCKS_PER_BATCH; ++k) {
                sum += epart[2 * (b * EDGE_BLOCKS_PER_BATCH + k) + 0];
                sq  += epart[2 * (b * EDGE_BLOCKS_PER_BATCH + k) + 1];
            }
            e_loss += (sq - sum * sum / (float)E_) / (float)(E_ - 1);
        }
        e_loss /= (float)B_;
        out[0] = W_CHAMFER * c_loss + W_EDGE * e_loss;
        out[1] = c_loss;
        out[2] = e_loss;
    }
}

extern "C" void kernel_launch(void* const* d_in, const int* in_sizes, int n_in,
                              void* d_out, int out_size, void* d_ws, size_t ws_size,
                              hipStream_t stream) {
    (void)in_sizes; (void)n_in; (void)out_size; (void)ws_size;
    const float* pred  = (const float*)d_in[0];
    const float* targ  = (const float*)d_in[1];
    const int*   edges = (const int*)d_in[2];
    float* out = (float*)d_out;

    char* ws = (char*)d_ws;
    unsigned* rowmin = (unsigned*)ws;                                        // B*N u32
    unsigned* colmin = (unsigned*)(ws + sizeof(unsigned) * (size_t)B_ * N_); // B*M u32
    float*    epart  = (float*)(ws + sizeof(unsigned) * ((size_t)B_ * N_ + (size_t)B_ * M_));

    init_ws_kernel<<<(B_ * M_ + 255) / 256, 256, 0, stream>>>(rowmin, colmin);
    dim3 grid(B_ * WGS_PER_BATCH, MSPLIT);
    chamfer_kernel<<<grid, 256, 0, stream>>>(pred, targ, rowmin, colmin);
    edge_kernel<<<(B_ * E_) / 256, 256, 0, stream>>>(pred, edges, epart);
    final_kernel<<<1, 256, 0, stream>>>(rowmin, colmin, epart, out);
}